// Interpolate4D_35235911696911
// MI455X (gfx1250) — compile-verified
//
#include <hip/hip_runtime.h>

// Quadrilinear interpolation [1,16,5,5,128,128] -> [1,16,8,8,256,256], f32.
// Memory-bound (256MB output write dominates). Strategy: stage the 4 (x,y)
// corner planes for a 32-uo-row stripe into LDS with CDNA5 async global->LDS
// DMA (ASYNCcnt), then do the 16-corner lerp from LDS with FMAs and write
// coalesced float4 stores.

namespace {

constexpr int kNRows = 18;                     // staged input U-rows per corner
constexpr int kVin   = 128;                    // input V length
constexpr int kLdsFloats = 4 * kNRows * kVin;  // 9216 floats = 36 KB

__device__ __forceinline__ float lerpf(float a, float b, float w) {
  return __builtin_fmaf(b - a, w, a);  // a + (b-a)*w, matches reference
}

#if defined(__HIP_DEVICE_COMPILE__)
typedef int v4i __attribute__((ext_vector_type(4)));
typedef __attribute__((address_space(1))) v4i* gptr_v4i;  // global (AS1)
typedef __attribute__((address_space(3))) v4i* lptr_v4i;  // LDS (AS3)
#endif

// 16-byte async copy global -> LDS (per-lane). CDNA5 GLOBAL_LOAD_ASYNC_TO_LDS_B128.
__device__ __forceinline__ void stage16(const float* __restrict__ g, float* l) {
#if defined(__HIP_DEVICE_COMPILE__)
 #if __has_builtin(__builtin_amdgcn_global_load_async_to_lds_b128)
  v4i* gg = (v4i*)const_cast<float*>(g);   // drop const, retype in generic AS
  v4i* ll = (v4i*)l;
  __builtin_amdgcn_global_load_async_to_lds_b128(
      (gptr_v4i)gg,          // explicit addrspacecast generic -> global
      (lptr_v4i)ll,          // explicit addrspacecast generic -> LDS
      /*offset=*/0, /*cpol=*/0);
 #else
  unsigned loff = (unsigned)(unsigned long long)
      (__attribute__((address_space(3))) void*)l;
  asm volatile("global_load_async_to_lds_b128 %0, %1, off"
               :: "v"(loff), "v"(g)
               : "memory");
 #endif
#else
  (void)g; (void)l;
#endif
}

__device__ __forceinline__ void wait_async0() {
#if defined(__HIP_DEVICE_COMPILE__)
 #if __has_builtin(__builtin_amdgcn_s_wait_asynccnt)
  __builtin_amdgcn_s_wait_asynccnt(0);
 #else
  asm volatile("s_wait_asynccnt 0" ::: "memory");
 #endif
#endif
}

}  // namespace

__global__ __launch_bounds__(256)
void Interpolate4D_kernel(const float* __restrict__ in, float* __restrict__ out) {
  __shared__ float smem[kLdsFloats];

  const int b      = blockIdx.x;      // 8192 blocks = 16c * 8x * 8y * 8 uo-stripes
  const int uo_blk = b & 7;
  const int yo     = (b >> 3) & 7;
  const int xo     = (b >> 6) & 7;
  const int c      = (b >> 9) & 15;
  const int tid    = threadIdx.x;

  const float sXY = 4.0f / 7.0f;      // (5-1)/(8-1), align_corners
  const float sUV = 127.0f / 255.0f;  // (128-1)/(256-1), align_corners

  // X / Y corner indices and weights
  float cx = (float)xo * sXY;
  int x0 = (int)cx; if (x0 > 4) x0 = 4;
  int x1 = x0 + 1;  if (x1 > 4) x1 = 4;
  float wx = cx - (float)x0;
  float cy = (float)yo * sXY;
  int y0 = (int)cy; if (y0 > 4) y0 = 4;
  int y1 = y0 + 1;  if (y1 > 4) y1 = 4;
  float wy = cy - (float)y0;

  const float cw[4] = { (1.f - wx) * (1.f - wy), (1.f - wx) * wy,
                        wx * (1.f - wy),         wx * wy };
  // corner slice base offsets (slice = 128*128 floats)
  const int cbase[4] = { ((c * 5 + x0) * 5 + y0) << 14,
                         ((c * 5 + x0) * 5 + y1) << 14,
                         ((c * 5 + x1) * 5 + y0) << 14,
                         ((c * 5 + x1) * 5 + y1) << 14 };

  const int uo_base = uo_blk * 32;
  const int u_base  = (int)((float)uo_base * sUV);  // first staged input U-row

  // ---- Stage: 4 corners x 18 rows x 128 floats -> LDS via async DMA ----
  // Per corner: 18*32 = 576 sixteen-byte chunks; 3 predicated rounds of 256.
#pragma unroll
  for (int corner = 0; corner < 4; ++corner) {
    const float* gb = in + cbase[corner];
    float* lb = &smem[corner * kNRows * kVin];
#pragma unroll
    for (int it = 0; it < 3; ++it) {
      int idx = it * 256 + tid;
      if (idx < kNRows * 32) {
        int j  = idx >> 5;        // staged row slot
        int v4 = idx & 31;        // 16B chunk within row
        int ru = u_base + j; if (ru > 127) ru = 127;  // clamp = edge replicate
        stage16(gb + ru * kVin + (v4 << 2), lb + (j << 7) + (v4 << 2));
      }
    }
  }
  wait_async0();
  __syncthreads();

  // ---- Compute: 32 uo rows x 256 vo, each thread does 8 rows x 4 vo ----
  const int vo0 = (tid & 63) << 2;  // float4 column start
  float* outp = out + ((size_t)((((c << 3) | xo) << 3) | yo) << 16);

#pragma unroll
  for (int p = 0; p < 8; ++p) {
    const int r  = p * 4 + (tid >> 6);
    const int uo = uo_base + r;
    float cu = (float)uo * sUV;
    int u0 = (int)cu; if (u0 > 127) u0 = 127;
    const float wu = cu - (float)u0;
    const int ju = u0 - u_base;     // 0..16; ju+1 <= 17 < kNRows

    float res[4];
#pragma unroll
    for (int q = 0; q < 4; ++q) {
      const int vo = vo0 + q;
      float cv = (float)vo * sUV;
      int v0 = (int)cv; if (v0 > 127) v0 = 127;
      int v1 = v0 + 1;  if (v1 > 127) v1 = 127;
      const float wv = cv - (float)v0;

      float acc = 0.0f;
#pragma unroll
      for (int cc = 0; cc < 4; ++cc) {
        const float* r0 = &smem[(cc * kNRows + ju) * kVin];
        const float* r1 = r0 + kVin;
        float a0 = lerpf(r0[v0], r0[v1], wv);   // V lerp, row u0
        float a1 = lerpf(r1[v0], r1[v1], wv);   // V lerp, row u0+1
        acc = __builtin_fmaf(cw[cc], lerpf(a0, a1, wu), acc);  // U lerp + XY blend
      }
      res[q] = acc;
    }
    float4 v4out = make_float4(res[0], res[1], res[2], res[3]);
    *(float4*)(outp + uo * 256 + vo0) = v4out;
  }
}

extern "C" void kernel_launch(void* const* d_in, const int* in_sizes, int n_in,
                              void* d_out, int out_size, void* d_ws, size_t ws_size,
                              hipStream_t stream) {
  (void)in_sizes; (void)n_in; (void)out_size; (void)d_ws; (void)ws_size;
  const float* in = (const float*)d_in[0];
  float* out = (float*)d_out;
  // 16c * 8x * 8y * 8 uo-stripes = 8192 blocks; 256 threads (8 waves) each.
  hipLaunchKernelGGL(Interpolate4D_kernel, dim3(8192), dim3(256), 0, stream, in, out);
}